// GATV2_17910013624718
// MI455X (gfx1250) — compile-verified
//
#include <hip/hip_runtime.h>
#include <hip/hip_bf16.h>

// ---------------------------------------------------------------------------
// GATv2 stack for MI455X (gfx1250, wave32).
// Dense node transforms use the native FP32 WMMA (V_WMMA_F32_16X16X4_F32):
// workload is memory/atomic bound (~20 FLOP/B vs 23.3 TB/s HBM), so full
// fp32 precision is free and the matrix pipe offloads the GEMMs from VALU.
// Edge passes: wave-per-edge, segment softmax via 3 passes with atomics.
// ---------------------------------------------------------------------------

typedef __attribute__((ext_vector_type(2))) float v2f;
typedef __attribute__((ext_vector_type(8))) float v8f;

#define WAVE 32

// ----- order-preserving float <-> uint encoding for atomicMax ---------------
__device__ __forceinline__ unsigned fenc(float f) {
    unsigned u = __float_as_uint(f);
    return (u & 0x80000000u) ? ~u : (u | 0x80000000u);
}
__device__ __forceinline__ float fdec(unsigned u) {
    u = (u & 0x80000000u) ? (u ^ 0x80000000u) : ~u;
    return __uint_as_float(u);
}

// ----- generic fill ---------------------------------------------------------
__global__ void fill_u32(unsigned* __restrict__ p, unsigned v, int n) {
    int t = blockIdx.x * blockDim.x + threadIdx.x;
    if (t < n) p[t] = v;
}

// ---------------------------------------------------------------------------
// FP32 WMMA GEMM: Out[M,Nc] = A[M,K] @ W[K,Nc].
// One wave computes one 16x16 tile; K stepped by 4 with V_WMMA_F32_16X16X4.
// A frag (16x4 f32, 2 VGPR): lanes 0-15 -> M=lane, K={k0,k0+1};
//                            lanes 16-31 -> M=lane-16, K={k0+2,k0+3}.
// B frag (4x16 f32, 2 VGPR): mirrored (N striped across lanes).
// C/D (16x16 f32, 8 VGPR):  vgpr i -> rows i and i+8 split by lane half.
// M (=20000) and Nc (128 or 16) are multiples of 16: no edge guards needed,
// EXEC stays all-ones around the WMMA as the ISA requires.
// ---------------------------------------------------------------------------
__global__ __launch_bounds__(WAVE)
void gemm_wmma_f32(const float* __restrict__ A, const float* __restrict__ W,
                   float* __restrict__ Out, int K, int Nc)
{
    const int lane = threadIdx.x & 31;
    const int half = lane >> 4;        // 0: K pair {0,1}, 1: K pair {2,3}
    const int l15  = lane & 15;
    const int rowBase = blockIdx.x * 16;
    const int colBase = blockIdx.y * 16;

    v8f acc = {};
    for (int k0 = 0; k0 < K; k0 += 4) {
        const int kb = k0 + half * 2;
        v2f a, b;
        const float* ap = A + (size_t)(rowBase + l15) * K + kb;
        a.x = ap[0];
        a.y = ap[1];
        const float* bp = W + (size_t)kb * Nc + colBase + l15;
        b.x = bp[0];
        b.y = bp[Nc];
        acc = __builtin_amdgcn_wmma_f32_16x16x4_f32(
            /*neg_a=*/false, a, /*neg_b=*/false, b,
            /*c_mod=*/(short)0, acc, /*reuse_a=*/false, /*reuse_b=*/false);
    }
#pragma unroll
    for (int i = 0; i < 8; ++i)
        Out[(size_t)(rowBase + half * 8 + i) * Nc + colBase + l15] = acc[i];
}

// ---------------------------------------------------------------------------
// Edge pass 1: score[e,h] = sum_c leakyrelu(xl[src]+xr[dst]+ea@We)[h,c]*att[h,c]
// One wave per edge. ee recomputed on the fly (16xTOT, weights L2-resident).
// atomicMax into encoded per-(dst,head) running max.
// ---------------------------------------------------------------------------
template<int H, int C>
__global__ __launch_bounds__(256)
void edge_score(const float* __restrict__ xl, const float* __restrict__ xr,
                const float* __restrict__ ea, const float* __restrict__ We,
                const float* __restrict__ att,
                const int* __restrict__ src, const int* __restrict__ dst,
                float* __restrict__ score, unsigned* __restrict__ smax, int E)
{
    constexpr int TOT = H * C;
    constexpr int CPL = (TOT + 31) / 32;   // channels per lane
    constexpr int LPH = C / CPL;           // lanes per head
    const int e    = (blockIdx.x * blockDim.x + threadIdx.x) >> 5;
    const int lane = threadIdx.x & 31;
    if (e >= E) return;
    const int s = src[e], d = dst[e];
    const float* eaE = ea + (size_t)e * 16;

    float ead[16];
#pragma unroll
    for (int dd = 0; dd < 16; ++dd) ead[dd] = eaE[dd];

    float sc = 0.f;
    const int c0 = lane * CPL;
    if (c0 < TOT) {
#pragma unroll
        for (int j = 0; j < CPL; ++j) {
            const int c = c0 + j;
            float ee = 0.f;
#pragma unroll
            for (int dd = 0; dd < 16; ++dd) ee += ead[dd] * We[dd * TOT + c];
            float g = xl[(size_t)s * TOT + c] + xr[(size_t)d * TOT + c] + ee;
            g = (g > 0.f) ? g : 0.2f * g;          // leaky_relu, slope 0.2
            sc += g * att[c];
        }
    }
#pragma unroll
    for (int off = 1; off < LPH; off <<= 1) sc += __shfl_xor(sc, off);
    if (c0 < TOT && (lane % LPH) == 0) {
        const int h = c0 / C;
        score[(size_t)e * H + h] = sc;
        atomicMax(&smax[(size_t)d * H + h], fenc(sc));
    }
}

// ---------------------------------------------------------------------------
// Edge pass 2: ex = exp(score - smax[dst]); score <- ex; den[dst] += ex
// ---------------------------------------------------------------------------
__global__ __launch_bounds__(256)
void edge_exp(const int* __restrict__ dst, float* __restrict__ score,
              const unsigned* __restrict__ smax, float* __restrict__ den,
              int E, int H)
{
    int t = blockIdx.x * blockDim.x + threadIdx.x;
    if (t >= E * H) return;
    const int e = t / H, h = t - e * H;
    const int d = dst[e];
    float ex = __expf(score[t] - fdec(smax[d * H + h]));
    score[t] = ex;
    atomicAdd(&den[d * H + h], ex);
}

// ---------------------------------------------------------------------------
// Edge pass 3: out[dst,h,c] += (ex / den[dst,h]) * xl[src,h,c]
// ---------------------------------------------------------------------------
template<int H, int C>
__global__ __launch_bounds__(256)
void edge_aggr(const float* __restrict__ xl, const float* __restrict__ score,
               const float* __restrict__ den,
               const int* __restrict__ src, const int* __restrict__ dst,
               float* __restrict__ out, int E)
{
    constexpr int TOT = H * C;
    constexpr int CPL = (TOT + 31) / 32;
    const int e    = (blockIdx.x * blockDim.x + threadIdx.x) >> 5;
    const int lane = threadIdx.x & 31;
    if (e >= E) return;
    const int s = src[e], d = dst[e];
    const int c0 = lane * CPL;
    if (c0 >= TOT) return;
#pragma unroll
    for (int j = 0; j < CPL; ++j) {
        const int c = c0 + j;
        const int h = c / C;
        const float alpha =
            score[(size_t)e * H + h] / (den[(size_t)d * H + h] + 1e-16f);
        atomicAdd(&out[(size_t)d * TOT + c], alpha * xl[(size_t)s * TOT + c]);
    }
}

// ---------------------------------------------------------------------------
// Conv1 epilogue: hx = elu(out1 + b) (in place); s_in = hx + xl1 (skip)
// ---------------------------------------------------------------------------
__global__ __launch_bounds__(256)
void conv1_post(float* __restrict__ out1, const float* __restrict__ xl1,
                const float* __restrict__ b, float* __restrict__ s_in, int total)
{
    int t = blockIdx.x * blockDim.x + threadIdx.x;
    if (t >= total) return;
    float h = out1[t] + b[t & 127];
    float hx = (h > 0.f) ? h : (__expf(h) - 1.f);     // elu
    out1[t] = hx;
    s_in[t] = hx + xl1[t];
}

// ---------------------------------------------------------------------------
// Final: x1 = out2+c2_b ; x2 = (outS+s_b)@lin_W^T + lin_b ; LayerNorm(x1+x2)
// 16 lanes per node; shfl reductions stay within the 16-lane group.
// ---------------------------------------------------------------------------
__global__ __launch_bounds__(256)
void final_kernel(const float* __restrict__ out2, const float* __restrict__ outS,
                  const float* __restrict__ c2_b, const float* __restrict__ s_b,
                  const float* __restrict__ lin_W, const float* __restrict__ lin_b,
                  const float* __restrict__ ln_g, const float* __restrict__ ln_b,
                  float* __restrict__ y, int N)
{
    int g = blockIdx.x * blockDim.x + threadIdx.x;
    const int n = g >> 4, c = g & 15;
    if (n >= N) return;
    const float x1 = out2[n * 16 + c] + c2_b[c];
    float acc = lin_b[c];
#pragma unroll
    for (int k = 0; k < 16; ++k)
        acc += (outS[n * 16 + k] + s_b[k]) * lin_W[c * 16 + k];
    float v = x1 + acc;
    float mu = v;
#pragma unroll
    for (int off = 1; off < 16; off <<= 1) mu += __shfl_xor(mu, off);
    mu *= (1.f / 16.f);
    float dv = (v - mu) * (v - mu);
#pragma unroll
    for (int off = 1; off < 16; off <<= 1) dv += __shfl_xor(dv, off);
    dv *= (1.f / 16.f);
    y[n * 16 + c] = (v - mu) * rsqrtf(dv + 1e-5f) * ln_g[c] + ln_b[c];
}

// ---------------------------------------------------------------------------
// Host-side launcher
// ---------------------------------------------------------------------------
static inline int cdiv(long a, int b) { return (int)((a + b - 1) / b); }

extern "C" void kernel_launch(void* const* d_in, const int* in_sizes, int n_in,
                              void* d_out, int out_size, void* d_ws, size_t ws_size,
                              hipStream_t stream)
{
    (void)in_sizes; (void)n_in; (void)out_size; (void)ws_size;
    constexpr int B = 4, N = 20000, E = 640000, F = 64;
    constexpr int H1 = 8, C1 = 16, TOT1 = H1 * C1;   // 128
    constexpr int C2 = 16;

    const float* x     = (const float*)d_in[0];
    const float* eattr = (const float*)d_in[1];
    const int*   eidx  = (const int*)  d_in[2];
    const float* c1_Wl = (const float*)d_in[3];
    const float* c1_Wr = (const float*)d_in[4];
    const float* c1_We = (const float*)d_in[5];
    const float* c1_at = (const float*)d_in[6];
    const float* c1_b  = (const float*)d_in[7];
    const float* c2_Wl = (const float*)d_in[8];
    const float* c2_Wr = (const float*)d_in[9];
    const float* c2_We = (const float*)d_in[10];
    const float* c2_at = (const float*)d_in[11];
    const float* c2_b  = (const float*)d_in[12];
    const float* s_Wl  = (const float*)d_in[13];
    const float* s_Wr  = (const float*)d_in[14];
    const float* s_We  = (const float*)d_in[15];
    const float* s_at  = (const float*)d_in[16];
    const float* s_b   = (const float*)d_in[17];
    const float* lin_W = (const float*)d_in[18];
    const float* lin_b = (const float*)d_in[19];
    const float* ln_g  = (const float*)d_in[20];
    const float* ln_b  = (const float*)d_in[21];
    float* out = (float*)d_out;

    // workspace layout (floats), reused across batches
    float* ws = (float*)d_ws;
    size_t o = 0;
    float*    xl1   = ws + o; o += (size_t)N * TOT1;
    float*    xr1   = ws + o; o += (size_t)N * TOT1;
    float*    out1  = ws + o; o += (size_t)N * TOT1;   // later holds hx
    float*    s_in  = ws + o; o += (size_t)N * TOT1;
    float*    sc1   = ws + o; o += (size_t)E * H1;
    unsigned* smax1 = (unsigned*)(ws + o); o += (size_t)N * H1;
    float*    den1  = ws + o; o += (size_t)N * H1;
    float*    xl2   = ws + o; o += (size_t)N * C2;
    float*    xr2   = ws + o; o += (size_t)N * C2;
    float*    out2  = ws + o; o += (size_t)N * C2;
    float*    xls   = ws + o; o += (size_t)N * C2;
    float*    xrs   = ws + o; o += (size_t)N * C2;
    float*    outS  = ws + o; o += (size_t)N * C2;
    float*    sc2   = ws + o; o += (size_t)E;
    unsigned* smax2 = (unsigned*)(ws + o); o += (size_t)N;
    float*    den2  = ws + o; o += (size_t)N;

    const unsigned ENC_NEG_INF = 0x007FFFFFu;  // fenc(-inf)
    const int EB = cdiv((long)E * WAVE, 256);  // wave-per-edge grids

    for (int b = 0; b < B; ++b) {
        const float* xb  = x     + (size_t)b * N * F;
        const float* eab = eattr + (size_t)b * E * 16;
        const int*   src = eidx  + (size_t)b * 2 * E;
        const int*   dst = src + E;

        // ---- conv1 (heads=8, ch=16, concat) ----
        gemm_wmma_f32<<<dim3(N / 16, TOT1 / 16), WAVE, 0, stream>>>(xb, c1_Wl, xl1, F, TOT1);
        gemm_wmma_f32<<<dim3(N / 16, TOT1 / 16), WAVE, 0, stream>>>(xb, c1_Wr, xr1, F, TOT1);
        fill_u32<<<cdiv((long)N * H1, 256), 256, 0, stream>>>(smax1, ENC_NEG_INF, N * H1);
        fill_u32<<<cdiv((long)N * H1, 256), 256, 0, stream>>>((unsigned*)den1, 0u, N * H1);
        fill_u32<<<cdiv((long)N * TOT1, 256), 256, 0, stream>>>((unsigned*)out1, 0u, N * TOT1);
        edge_score<H1, C1><<<EB, 256, 0, stream>>>(xl1, xr1, eab, c1_We, c1_at, src, dst, sc1, smax1, E);
        edge_exp<<<cdiv((long)E * H1, 256), 256, 0, stream>>>(dst, sc1, smax1, den1, E, H1);
        edge_aggr<H1, C1><<<EB, 256, 0, stream>>>(xl1, sc1, den1, src, dst, out1, E);
        conv1_post<<<cdiv((long)N * TOT1, 256), 256, 0, stream>>>(out1, xl1, c1_b, s_in, N * TOT1);
        // out1 now holds hx = elu(conv1 out); s_in = hx + skip

        // ---- conv2 (heads=1, ch=16) on hx ----
        gemm_wmma_f32<<<dim3(N / 16, C2 / 16), WAVE, 0, stream>>>(out1, c2_Wl, xl2, TOT1, C2);
        gemm_wmma_f32<<<dim3(N / 16, C2 / 16), WAVE, 0, stream>>>(out1, c2_Wr, xr2, TOT1, C2);
        fill_u32<<<cdiv(N, 256), 256, 0, stream>>>(smax2, ENC_NEG_INF, N);
        fill_u32<<<cdiv(N, 256), 256, 0, stream>>>((unsigned*)den2, 0u, N);
        fill_u32<<<cdiv((long)N * C2, 256), 256, 0, stream>>>((unsigned*)out2, 0u, N * C2);
        edge_score<1, C2><<<EB, 256, 0, stream>>>(xl2, xr2, eab, c2_We, c2_at, src, dst, sc2, smax2, E);
        edge_exp<<<cdiv(E, 256), 256, 0, stream>>>(dst, sc2, smax2, den2, E, 1);
        edge_aggr<1, C2><<<EB, 256, 0, stream>>>(xl2, sc2, den2, src, dst, out2, E);

        // ---- skip conv (heads=1, ch=16) on hx + skip ----
        gemm_wmma_f32<<<dim3(N / 16, C2 / 16), WAVE, 0, stream>>>(s_in, s_Wl, xls, TOT1, C2);
        gemm_wmma_f32<<<dim3(N / 16, C2 / 16), WAVE, 0, stream>>>(s_in, s_Wr, xrs, TOT1, C2);
        fill_u32<<<cdiv(N, 256), 256, 0, stream>>>(smax2, ENC_NEG_INF, N);
        fill_u32<<<cdiv(N, 256), 256, 0, stream>>>((unsigned*)den2, 0u, N);
        fill_u32<<<cdiv((long)N * C2, 256), 256, 0, stream>>>((unsigned*)outS, 0u, N * C2);
        edge_score<1, C2><<<EB, 256, 0, stream>>>(xls, xrs, eab, s_We, s_at, src, dst, sc2, smax2, E);
        edge_exp<<<cdiv(E, 256), 256, 0, stream>>>(dst, sc2, smax2, den2, E, 1);
        edge_aggr<1, C2><<<EB, 256, 0, stream>>>(xls, sc2, den2, src, dst, outS, E);

        // ---- linear skip + LayerNorm ----
        final_kernel<<<cdiv((long)N * 16, 256), 256, 0, stream>>>(
            out2, outS, c2_b, s_b, lin_W, lin_b, ln_g, ln_b,
            out + (size_t)b * N * 16, N);
    }
}